// RetinaNet_28200755266079
// MI455X (gfx1250) — compile-verified
//
#include <hip/hip_runtime.h>
#include <cstdint>

typedef float v2f __attribute__((ext_vector_type(2)));
typedef float v8f __attribute__((ext_vector_type(8)));
typedef unsigned long long u64;

#define NCLS     80
#define M_TOT    4900
#define NW       160            // 32-bit words per suppression row (5120 bits)
#define SP       5120           // padded candidate stride
#define NBIN     4096
#define RBLK     307            // 307*16 = 4912 rows of bit matrix
#define CONF_KEY 0x3D4CCCCDu    // bit pattern of 0.05f

// workspace offsets in u32 units
enum : int {
  OFF_HIST   = 0,        // 5*4096
  OFF_HIST2  = 20480,    // 5*4096
  OFF_CNT    = 40960,    // 64 counters/state
  OFF_CKEY   = 41024,    // SP
  OFF_CPRIO  = 46144,    // SP
  OFF_SSCORE = 51264,    // SP
  OFF_SLABEL = 56384,    // SP
  OFF_SBOX   = 61504,    // SP*4
  OFF_AREA   = 81984,    // SP
  OFF_OBOX   = 87104,    // SP*4
  OFF_BITS   = 107584,   // 4912*160
};
enum : int { C_HI=0, C_EQ=5, C_MID=10, C_CUT1=15, C_AB1=20, C_QU1=25,
             C_CUT2=30, C_AB2=35, C_QU2=40 };

__device__ __forceinline__ unsigned fkey(float f) {
  unsigned u = __float_as_uint(f);
  return (u & 0x80000000u) ? ~u : (u | 0x80000000u);   // monotone float->uint
}

__device__ __forceinline__ void wait_async0() {
#if __has_builtin(__builtin_amdgcn_s_wait_asynccnt)
  __builtin_amdgcn_s_wait_asynccnt(0);
#else
  asm volatile("s_wait_asynccnt 0" ::: "memory");
#endif
}

// each lane DMAs 16B from global into LDS (tracked by ASYNCcnt)
__device__ __forceinline__ void async_tile16(const float* g, float* lds) {
  const unsigned la = (unsigned)(size_t)lds;
  const unsigned long long ga = (unsigned long long)(size_t)g;
  asm volatile("global_load_async_to_lds_b128 %0, %1, off"
               :: "v"(la), "v"(ga) : "memory");
}

// ---------------- init ----------------
__global__ void k_init(unsigned* wsu) {
  const int n = OFF_CNT + 64;
  for (int i = blockIdx.x * blockDim.x + threadIdx.x; i < n;
       i += gridDim.x * blockDim.x)
    wsu[i] = 0u;
}

// ------------- pass 1 histogram: async global->LDS streaming -------------
__global__ void k_hist1(const float* __restrict__ cls, int n,
                        unsigned* __restrict__ ghist) {
  __shared__ unsigned lh[NBIN];
  __shared__ float tile[1024];
  const int tid = threadIdx.x;                 // 256 threads
  for (int i = tid; i < NBIN; i += blockDim.x) lh[i] = 0u;
  __syncthreads();
  const int ntiles = (n + 1023) >> 10;
  for (int t = blockIdx.x; t < ntiles; t += gridDim.x) {
    const int base = t << 10;
    if (n - base >= 1024) {
      async_tile16(cls + base + tid * 4, &tile[tid * 4]);
      wait_async0();
      __syncthreads();
#pragma unroll
      for (int u = 0; u < 4; ++u) {
        const unsigned key = fkey(tile[tid + u * 256]);
        atomicAdd(&lh[key >> 20], 1u);
      }
      __syncthreads();
    } else {
      for (int i = base + tid; i < n; i += blockDim.x)
        atomicAdd(&lh[fkey(cls[i]) >> 20], 1u);
    }
  }
  __syncthreads();
  for (int i = tid; i < NBIN; i += blockDim.x) {
    const unsigned c = lh[i];
    if (c) atomicAdd(&ghist[i], c);
  }
}

// ------------- select level-1 cutoff -------------
__global__ void k_select1(unsigned* wsu) {
  if (threadIdx.x != 0) return;
  const int lvl = blockIdx.x;
  const unsigned K = (lvl == 4) ? 900u : 1000u;
  const unsigned* h = wsu + OFF_HIST + lvl * NBIN;
  unsigned cum = 0; int b = NBIN - 1;
  for (; b > 0; --b) { const unsigned c = h[b]; if (cum + c >= K) break; cum += c; }
  wsu[OFF_CNT + C_CUT1 + lvl] = (unsigned)b;
  wsu[OFF_CNT + C_AB1  + lvl] = cum;
  wsu[OFF_CNT + C_QU1  + lvl] = K - cum;
}

// ------------- pass 2 histogram (refine within cutoff bin; input is L2-resident) -------------
__global__ void k_hist2(const float* __restrict__ cls, int n, int lvl,
                        unsigned* __restrict__ wsu) {
  __shared__ unsigned lh[NBIN];
  __shared__ float tile[1024];
  const int tid = threadIdx.x;                 // 256 threads
  for (int i = tid; i < NBIN; i += blockDim.x) lh[i] = 0u;
  __syncthreads();
  const unsigned cut1 = wsu[OFF_CNT + C_CUT1 + lvl];
  const int ntiles = (n + 1023) >> 10;
  for (int t = blockIdx.x; t < ntiles; t += gridDim.x) {
    const int base = t << 10;
    if (n - base >= 1024) {
      async_tile16(cls + base + tid * 4, &tile[tid * 4]);
      wait_async0();
      __syncthreads();
#pragma unroll
      for (int u = 0; u < 4; ++u) {
        const unsigned key = fkey(tile[tid + u * 256]);
        if ((key >> 20) == cut1) atomicAdd(&lh[(key >> 8) & 0xFFFu], 1u);
      }
      __syncthreads();
    } else {
      for (int i = base + tid; i < n; i += blockDim.x) {
        const unsigned key = fkey(cls[i]);
        if ((key >> 20) == cut1) atomicAdd(&lh[(key >> 8) & 0xFFFu], 1u);
      }
    }
  }
  __syncthreads();
  unsigned* gh = wsu + OFF_HIST2 + lvl * NBIN;
  for (int i = tid; i < NBIN; i += blockDim.x) {
    const unsigned c = lh[i];
    if (c) atomicAdd(&gh[i], c);
  }
}

__global__ void k_select2(unsigned* wsu) {
  if (threadIdx.x != 0) return;
  const int lvl = blockIdx.x;
  const unsigned K = wsu[OFF_CNT + C_QU1 + lvl];
  const unsigned* h = wsu + OFF_HIST2 + lvl * NBIN;
  unsigned cum = 0; int b = NBIN - 1;
  for (; b > 0; --b) { const unsigned c = h[b]; if (cum + c >= K) break; cum += c; }
  wsu[OFF_CNT + C_CUT2 + lvl] = (unsigned)b;
  wsu[OFF_CNT + C_AB2  + lvl] = cum;
  wsu[OFF_CNT + C_QU2  + lvl] = K - cum;
}

// ------------- compact top-k candidates per level -------------
__global__ void k_compact(const float* __restrict__ cls, int n, int lvl,
                          int base, unsigned* __restrict__ wsu) {
  const unsigned cut1 = wsu[OFF_CNT + C_CUT1 + lvl];
  const unsigned ab1  = wsu[OFF_CNT + C_AB1  + lvl];
  const unsigned cut2 = wsu[OFF_CNT + C_CUT2 + lvl];
  const unsigned ab2  = wsu[OFF_CNT + C_AB2  + lvl];
  const unsigned qu2  = wsu[OFF_CNT + C_QU2  + lvl];
  for (int i = blockIdx.x * blockDim.x + threadIdx.x; i < n;
       i += gridDim.x * blockDim.x) {
    const float logit = cls[i];
    const unsigned key = fkey(logit);
    const unsigned b1 = key >> 20;
    int slot = -1;
    if (b1 > cut1) {
      slot = (int)atomicAdd(&wsu[OFF_CNT + C_HI + lvl], 1u);
    } else if (b1 == cut1) {
      const unsigned b2 = (key >> 8) & 0xFFFu;
      if (b2 > cut2) {
        slot = (int)(ab1 + atomicAdd(&wsu[OFF_CNT + C_MID + lvl], 1u));
      } else if (b2 == cut2) {
        const unsigned e = atomicAdd(&wsu[OFF_CNT + C_EQ + lvl], 1u);
        if (e < qu2) slot = (int)(ab1 + ab2 + e);
      }
    }
    if (slot >= 0) {
      const int g = base + slot;
      const float s = 1.0f / (1.0f + __expf(-logit));
      wsu[OFF_CKEY  + g] = __float_as_uint(s);          // score >= 0 -> uint order ok
      wsu[OFF_CPRIO + g] = ((unsigned)lvl << 25) | (unsigned)i;
    }
  }
}

// ------------- global bitonic sort of all 4900 candidates (LDS 96KB) -------------
__global__ void k_sort(unsigned* __restrict__ wsu, float* __restrict__ wsf,
                       const float* b0, const float* b1, const float* b2,
                       const float* b3, const float* b4) {
  extern __shared__ char sm[];
  u64* key      = (u64*)sm;                   // 8192 * 8B
  unsigned* pay = (unsigned*)(sm + 8192 * sizeof(u64));
  const int tid = threadIdx.x;                // 1024 threads
  for (int i = tid; i < 8192; i += blockDim.x) {
    u64 k = 0; unsigned p = 0xFFFFFFFFu;
    if (i < M_TOT) {
      const unsigned rk = wsu[OFF_CKEY + i];
      const unsigned mk = (rk > CONF_KEY) ? rk : 0u;    // masked score (invalid -> 0)
      k = ((u64)mk << 32) | (u64)rk;                    // masked major, raw minor
      p = wsu[OFF_CPRIO + i];
    }
    key[i] = k; pay[i] = p;
  }
  __syncthreads();
  for (int kk = 2; kk <= 8192; kk <<= 1) {
    for (int j = kk >> 1; j > 0; j >>= 1) {
      for (int i = tid; i < 8192; i += blockDim.x) {
        const int ixj = i ^ j;
        if (ixj > i) {
          const u64 ka = key[i], kb = key[ixj];
          const unsigned pa = pay[i], pb = pay[ixj];
          const bool desc = ((i & kk) == 0);
          const bool sw = desc ? ((ka < kb) || (ka == kb && pa > pb))
                               : ((ka > kb) || (ka == kb && pa < pb));
          if (sw) { key[i] = kb; key[ixj] = ka; pay[i] = pb; pay[ixj] = pa; }
        }
      }
      __syncthreads();
    }
  }
  const float* bps[5] = {b0, b1, b2, b3, b4};
  for (int i = tid; i < SP; i += blockDim.x) {
    float sc = 0.f; unsigned lb = 0u;
    float x1 = 0.f, y1 = 0.f, x2 = 0.f, y2 = 0.f;
    if (i < M_TOT) {
      const u64 k = key[i];
      const unsigned p = pay[i];
      sc = __uint_as_float((unsigned)(k >> 32));
      const unsigned lvl  = p >> 25;
      const unsigned elem = p & 0x1FFFFFFu;
      lb = elem % NCLS;
      const float* bp = bps[lvl] + (size_t)(elem / NCLS) * 4;
      x1 = bp[0]; y1 = bp[1]; x2 = bp[2]; y2 = bp[3];
    }
    wsf[OFF_SSCORE + i] = sc;
    wsu[OFF_SLABEL + i] = lb;
    wsf[OFF_SBOX + i * 4 + 0] = x1;
    wsf[OFF_SBOX + i * 4 + 1] = y1;
    wsf[OFF_SBOX + i * 4 + 2] = x2;
    wsf[OFF_SBOX + i * 4 + 3] = y2;
    const float off = (float)lb * 1281.0f;    // class-offset trick (IMG+1)
    wsf[OFF_OBOX + i * 4 + 0] = x1 + off;
    wsf[OFF_OBOX + i * 4 + 1] = y1 + off;
    wsf[OFF_OBOX + i * 4 + 2] = x2 + off;
    wsf[OFF_OBOX + i * 4 + 3] = y2 + off;
    wsf[OFF_AREA + i] = fmaxf(x2 - x1, 0.f) * fmaxf(y2 - y1, 0.f);
  }
}

// ------------- pairwise IoU suppression bits; WMMA computes area_i + area_j -------------
__device__ __forceinline__ float4 ldbox4(const float* p, int i) {
  return *(const float4*)(p + (size_t)i * 4);
}
__device__ __forceinline__ float interA(float4 a, float4 b) {
  const float lx = fmaxf(a.x, b.x), ly = fmaxf(a.y, b.y);
  const float rx = fminf(a.z, b.z), ry = fminf(a.w, b.w);
  return fmaxf(rx - lx, 0.f) * fmaxf(ry - ly, 0.f);
}
__device__ __forceinline__ unsigned jmask(int row, int j0) {
  // bits b: j=j0+b must satisfy j > row && j < M_TOT
  const int l = row - j0;
  if (l >= 31) return 0u;
  unsigned m = (l >= 0) ? (0xFFFFFFFFu << (l + 1)) : 0xFFFFFFFFu;
  const int r = M_TOT - j0;
  if (r <= 0) return 0u;
  if (r < 32) m &= (1u << r) - 1u;
  return m;
}

__global__ void k_iou_bits(const float* __restrict__ wsf,
                           unsigned* __restrict__ wsu) {
  const int lane   = threadIdx.x & 31;
  const int wave   = (blockIdx.x * blockDim.x + threadIdx.x) >> 5;
  const int nwaves = (gridDim.x * blockDim.x) >> 5;
  const float* area = wsf + OFF_AREA;
  const float* obox = wsf + OFF_OBOX;
  unsigned* bits = wsu + OFF_BITS;
  const bool lo = lane < 16;
  const int lm = lane & 15;
  const int ntask = RBLK * NW;                // each wave-task: 16 rows x 32 cols
  for (int task = wave; task < ntask; task += nwaves) {
    const int rb = task / NW;
    const int w  = task - rb * NW;
    const int r0 = rb * 16, c0 = w * 32;
    // A (16x4 f32): col0 = area_i, col1 = 1 ; B (4x16): row0 = 1, row1 = area_j
    v2f a;  a.x  = lo ? area[r0 + lm] : 0.f;       a.y  = lo ? 1.f : 0.f;
    v2f bA; bA.x = lo ? 1.f : 0.f;                 bA.y = lo ? area[c0 + lm] : 0.f;
    v2f bB; bB.x = bA.x;                           bB.y = lo ? area[c0 + 16 + lm] : 0.f;
    const v8f z = {0.f, 0.f, 0.f, 0.f, 0.f, 0.f, 0.f, 0.f};
    // C[m][n] = area_{r0+m} + area_{c0(+16)+n}
    v8f s0 = __builtin_amdgcn_wmma_f32_16x16x4_f32(false, a, false, bA,
                                                   (short)0, z, false, false);
    v8f s1 = __builtin_amdgcn_wmma_f32_16x16x4_f32(false, a, false, bB,
                                                   (short)0, z, false, false);
    const float4 bj0 = ldbox4(obox, c0 + lm);
    const float4 bj1 = ldbox4(obox, c0 + 16 + lm);
    const int ibase = r0 + (lo ? 0 : 8);
#pragma unroll
    for (int v = 0; v < 8; ++v) {
      const float4 bi = ldbox4(obox, ibase + v);
      const float in0 = interA(bi, bj0);
      const float in1 = interA(bi, bj1);
      // iou > T  <=>  inter > T*(sumArea - inter + eps)   (denominator > 0 always)
      const bool p0 = in0 > 0.6f * (s0[v] - in0 + 1e-9f);
      const bool p1 = in1 > 0.6f * (s1[v] - in1 + 1e-9f);
      const unsigned bal0 = __builtin_amdgcn_ballot_w32(p0);
      const unsigned bal1 = __builtin_amdgcn_ballot_w32(p1);
      if (lane == 0) {
        const int rA = r0 + v, rB = r0 + v + 8;
        const unsigned wA = (bal0 & 0xFFFFu) | ((bal1 & 0xFFFFu) << 16);
        const unsigned wB = ((bal0 >> 16) & 0xFFFFu) | (bal1 & 0xFFFF0000u);
        bits[(size_t)rA * NW + w] = wA & jmask(rA, c0);
        bits[(size_t)rB * NW + w] = wB & jmask(rB, c0);
      }
    }
  }
}

// ------------- serial greedy NMS over LDS bitmask + finalize outputs -------------
__global__ void k_nms_finalize(const float* __restrict__ wsf,
                               const unsigned* __restrict__ wsu,
                               float* __restrict__ out) {
  __shared__ unsigned keepm[NW];
  const int tid = threadIdx.x;                 // 256 threads
  for (int w = tid; w < NW; w += blockDim.x) {
    unsigned m = 0u;
    const int j0 = w * 32;
    for (int b = 0; b < 32; ++b) {
      const int j = j0 + b;
      if (j < M_TOT && wsf[OFF_SSCORE + j] > 0.05f) m |= (1u << b);
    }
    keepm[w] = m;
  }
  __syncthreads();
  const unsigned* bits = wsu + OFF_BITS;
  for (int i = 0; i < M_TOT; ++i) {
    const bool alive = (keepm[i >> 5] >> (i & 31)) & 1u;
    if (alive) {
      const unsigned* row = bits + (size_t)i * NW;
      for (int w = tid; w < NW; w += blockDim.x) keepm[w] &= ~row[w];
    }
    __syncthreads();
  }
  for (int i = tid; i < M_TOT; i += blockDim.x) {
    const float kf = ((keepm[i >> 5] >> (i & 31)) & 1u) ? 1.f : 0.f;
    const float inv = 1.0f / 1280.0f;
#pragma unroll
    for (int c = 0; c < 4; ++c) {
      const float v = wsf[OFF_SBOX + i * 4 + c] * inv;
      out[i * 4 + c] = fminf(fmaxf(v, 0.f), 1.f) * kf;
    }
    out[19600 + i] = wsf[OFF_SSCORE + i] * kf;           // scores
    out[24500 + i] = (float)wsu[OFF_SLABEL + i];         // labels (unmasked)
    out[29400 + i] = kf;                                 // keep
  }
}

extern "C" void kernel_launch(void* const* d_in, const int* in_sizes, int n_in,
                              void* d_out, int out_size, void* d_ws,
                              size_t ws_size, hipStream_t stream) {
  (void)in_sizes; (void)n_in; (void)out_size; (void)ws_size;
  unsigned* wsu = (unsigned*)d_ws;
  float* wsf = (float*)d_ws;
  float* out = (float*)d_out;
  static const int LN[5] = {230400, 57600, 14400, 3600, 900};

  k_init<<<161, 256, 0, stream>>>(wsu);
  for (int l = 0; l < 5; ++l) {
    const float* cls = (const float*)d_in[2 * l];
    k_hist1<<<1024, 256, 0, stream>>>(cls, LN[l] * NCLS,
                                      wsu + OFF_HIST + l * NBIN);
  }
  k_select1<<<5, 32, 0, stream>>>(wsu);
  for (int l = 0; l < 5; ++l) {
    const float* cls = (const float*)d_in[2 * l];
    k_hist2<<<1024, 256, 0, stream>>>(cls, LN[l] * NCLS, l, wsu);
  }
  k_select2<<<5, 32, 0, stream>>>(wsu);
  for (int l = 0; l < 5; ++l) {
    const float* cls = (const float*)d_in[2 * l];
    k_compact<<<1024, 256, 0, stream>>>(cls, LN[l] * NCLS, l, l * 1000, wsu);
  }
  k_sort<<<1, 1024, 8192 * 12, stream>>>(wsu, wsf,
                                         (const float*)d_in[1],
                                         (const float*)d_in[3],
                                         (const float*)d_in[5],
                                         (const float*)d_in[7],
                                         (const float*)d_in[9]);
  k_iou_bits<<<2048, 256, 0, stream>>>(wsf, wsu);
  k_nms_finalize<<<1, 256, 0, stream>>>(wsf, wsu, out);
}